// ModelDeepKernel_5317169512782
// MI455X (gfx1250) — compile-verified
//
#include <hip/hip_runtime.h>
#include <math.h>

typedef __attribute__((ext_vector_type(2))) float v2f;
typedef __attribute__((ext_vector_type(8))) float v8f;

#define N_PTS   8192
#define N_BATCH 4
#define NPOINT  409          // int(8192 * 0.05)
#define NS_REP  20
#define R2_REP  (0.07f * 0.07f)

// ---------------------------------------------------------------------------
// sorted insert of x into ascending best-5 (keeps the 5 smallest seen)
// ---------------------------------------------------------------------------
__device__ __forceinline__ void insert5(float x, float b[5]) {
#pragma unroll
  for (int k = 0; k < 5; ++k) {
    float lo = fminf(x, b[k]);
    float hi = fmaxf(x, b[k]);
    b[k] = lo;
    x = hi;
  }
}

// ---------------------------------------------------------------------------
// Farthest point sampling: 1 block per batch, 256 threads, 32 points/thread.
// Emits gathered coordinates new_xyz[b][it][3]. Matches jax scan semantics:
// emit current 'far' (starting at 0), update min-dists, argmax (first index).
// ---------------------------------------------------------------------------
__global__ __launch_bounds__(256) void fps_kernel(const float* __restrict__ pcd,
                                                  float* __restrict__ new_xyz) {
  __shared__ float redv[256];
  __shared__ int   redi[256];
  __shared__ float cpt[3];

  const int b   = blockIdx.x;
  const int tid = threadIdx.x;
  const float* P = pcd + (size_t)b * N_PTS * 3;

  float dmin[32];
#pragma unroll
  for (int k = 0; k < 32; ++k) dmin[k] = 1e10f;

  int far = 0;
  for (int it = 0; it < NPOINT; ++it) {
    if (tid == 0) {
      float fx = P[far * 3 + 0], fy = P[far * 3 + 1], fz = P[far * 3 + 2];
      float* o = new_xyz + ((size_t)b * NPOINT + it) * 3;
      o[0] = fx; o[1] = fy; o[2] = fz;
      cpt[0] = fx; cpt[1] = fy; cpt[2] = fz;
    }
    __syncthreads();
    const float fx = cpt[0], fy = cpt[1], fz = cpt[2];

    float bestv = -1.0f;
    int   besti = 0;
#pragma unroll 4
    for (int k = 0; k < 32; ++k) {
      const int i = tid + k * 256;
      float dx = P[i * 3 + 0] - fx;
      float dy = P[i * 3 + 1] - fy;
      float dz = P[i * 3 + 2] - fz;
      float dd = fmaf(dx, dx, fmaf(dy, dy, dz * dz));
      float nd = fminf(dmin[k], dd);
      dmin[k] = nd;
      if (nd > bestv) { bestv = nd; besti = i; }   // strict > keeps lowest index
    }
    redv[tid] = bestv;
    redi[tid] = besti;
    __syncthreads();
#pragma unroll
    for (int s = 128; s > 0; s >>= 1) {
      if (tid < s) {
        float v2 = redv[tid + s];
        int   i2 = redi[tid + s];
        if (v2 > redv[tid] || (v2 == redv[tid] && i2 < redi[tid])) {
          redv[tid] = v2; redi[tid] = i2;
        }
      }
      __syncthreads();
    }
    far = redi[0];
    __syncthreads();
  }
}

// ---------------------------------------------------------------------------
// Uniform loss, one percentage per launch. One wave per query (G = 4*409).
// Ordered ballquery compaction into wave-private LDS via ballot/popcount,
// pad with first hit, then nearest-non-self over the group.
// ---------------------------------------------------------------------------
__global__ __launch_bounds__(128) void uniform_kernel(
    const float* __restrict__ pcd, const float* __restrict__ new_xyz,
    float* __restrict__ out, int ns, float r2, float expect, float scale) {
  __shared__ float grp[4][132 * 3];

  const int wave = threadIdx.x >> 5;
  const int lane = threadIdx.x & 31;
  const int g = blockIdx.x * 4 + wave;
  if (g >= N_BATCH * NPOINT) return;

  const int b = g / NPOINT;
  const float* P = pcd + (size_t)b * N_PTS * 3;
  const float qx = new_xyz[g * 3 + 0];
  const float qy = new_xyz[g * 3 + 1];
  const float qz = new_xyz[g * 3 + 2];
  float* G = grp[wave];

  const unsigned below = (lane == 31) ? 0x7fffffffu : ((1u << lane) - 1u);
  int cnt = 0;
  for (int j0 = 0; j0 < N_PTS && cnt < ns; j0 += 32) {
    const int j = j0 + lane;
    const float px = P[j * 3 + 0], py = P[j * 3 + 1], pz = P[j * 3 + 2];
    const float dx = px - qx, dy = py - qy, dz = pz - qz;
    const float dd = fmaf(dx, dx, fmaf(dy, dy, dz * dz));
    const bool hit = (dd <= r2);
    const unsigned mask = __builtin_amdgcn_ballot_w32(hit);
    const int pos = cnt + __popc(mask & below);
    if (hit && pos < ns) {
      G[pos * 3 + 0] = px; G[pos * 3 + 1] = py; G[pos * 3 + 2] = pz;
    }
    cnt += __popc(mask);
    __builtin_amdgcn_wave_barrier();   // keep LDS ops ordered (wave-private)
  }
  if (cnt > ns) cnt = ns;

  __builtin_amdgcn_wave_barrier();
  const float f0x = G[0], f0y = G[1], f0z = G[2];   // first hit (self always hits)
  for (int k = cnt + lane; k < ns; k += 32) {
    G[k * 3 + 0] = f0x; G[k * 3 + 1] = f0y; G[k * 3 + 2] = f0z;
  }
  __builtin_amdgcn_wave_barrier();

  // nearest non-self squared distance per group member
  float acc = 0.0f;
  for (int i = lane; i < ns; i += 32) {
    const float ix = G[i * 3 + 0], iy = G[i * 3 + 1], iz = G[i * 3 + 2];
    float nn = 1e30f;
    for (int j = 0; j < ns; ++j) {
      const float dx = G[j * 3 + 0] - ix;
      const float dy = G[j * 3 + 1] - iy;
      const float dz = G[j * 3 + 2] - iz;
      const float dd = fmaf(dx, dx, fmaf(dy, dy, dz * dz));
      nn = (j == i) ? nn : fminf(nn, dd);
    }
    const float u = sqrtf(nn + 1e-8f);
    const float t = u - expect;
    acc += t * t;
  }
  acc *= scale;
#pragma unroll
  for (int off = 16; off; off >>= 1) acc += __shfl_down(acc, off);
  if (lane == 0) atomicAdd(&out[0], acc);
}

// ---------------------------------------------------------------------------
// Repulsion loss. 256 threads = 8 waves, 32 queries per wave (two A tiles).
// Distance tiles via V_WMMA_F32_16X16X4_F32:  d = |q|^2 + |p|^2 - 2 q.p.
// A(16x4)/B(4x16) f32 operand layout: lane<16 -> (x,y) of elem (lane&15),
// lane>=16 -> (z,0).  D layout: VGPR v = row (v | v+8), col = lane&15.
// All LDS traffic is wave-private: LDS is in-order per wave (DScnt), so no
// s_barrier is needed -- only compiler scheduling fences.
// ---------------------------------------------------------------------------
__global__ __launch_bounds__(256) void repulsion_kernel(const float* __restrict__ pcd,
                                                        float* __restrict__ out) {
  __shared__ __align__(16) float distS[8][32 * 20];  // [wave][query][db], stride 20
  __shared__ float qnS[8][32];

  const int tid  = threadIdx.x;
  const int wave = tid >> 5;
  const int lane = tid & 31;
  const int col  = lane & 15;
  const bool hi  = (lane >= 16);

  const int qglob = blockIdx.x * 256 + wave * 32;   // over B*N
  const int b  = qglob >> 13;
  const int q0 = qglob & (N_PTS - 1);
  const float* P = pcd + (size_t)b * N_PTS * 3;

  // per-lane query norm (query q0+lane)
  {
    const float x = P[(q0 + lane) * 3 + 0];
    const float y = P[(q0 + lane) * 3 + 1];
    const float z = P[(q0 + lane) * 3 + 2];
    qnS[wave][lane] = fmaf(x, x, fmaf(y, y, z * z));
  }

  // A operands: tile0 = queries q0..q0+15, tile1 = q0+16..q0+31
  v2f A0, A1;
  {
    const float x0 = P[(q0 + col) * 3 + 0];
    const float y0 = P[(q0 + col) * 3 + 1];
    const float z0 = P[(q0 + col) * 3 + 2];
    A0.x = hi ? z0 : x0;
    A0.y = hi ? 0.0f : y0;
    const float x1 = P[(q0 + 16 + col) * 3 + 0];
    const float y1 = P[(q0 + 16 + col) * 3 + 1];
    const float z1 = P[(q0 + 16 + col) * 3 + 2];
    A1.x = hi ? z1 : x1;
    A1.y = hi ? 0.0f : y1;
  }
  __builtin_amdgcn_wave_barrier();   // qnS cross-lane read below (same wave)

  // query norms for the rows this lane produces
  float qr0[8], qr1[8];
#pragma unroll
  for (int v = 0; v < 8; ++v) {
    const int m = v + (hi ? 8 : 0);
    qr0[v] = qnS[wave][m];
    qr1[v] = qnS[wave][16 + m];
  }

  float best[5];
#pragma unroll
  for (int k = 0; k < 5; ++k) best[k] = 1e30f;
  float firstd = 0.0f;
  int cnt = 0;

  const float4* row = reinterpret_cast<const float4*>(&distS[wave][lane * 20]);

  for (int tb = 0; tb < N_PTS / 16; ++tb) {
    const int j0 = tb * 16;
    // B operand: db point j0+col
    const float bx = P[(j0 + col) * 3 + 0];
    const float by = P[(j0 + col) * 3 + 1];
    const float bz = P[(j0 + col) * 3 + 2];
    v2f Bm;
    Bm.x = hi ? bz : bx;
    Bm.y = hi ? 0.0f : by;
    const float pn = fmaf(bx, bx, fmaf(by, by, bz * bz));

    v8f s0 = {0.f, 0.f, 0.f, 0.f, 0.f, 0.f, 0.f, 0.f};
    v8f s1 = {0.f, 0.f, 0.f, 0.f, 0.f, 0.f, 0.f, 0.f};
    s0 = __builtin_amdgcn_wmma_f32_16x16x4_f32(false, A0, false, Bm, (short)0, s0,
                                               false, false);
    s1 = __builtin_amdgcn_wmma_f32_16x16x4_f32(false, A1, false, Bm, (short)0, s1,
                                               false, false);

#pragma unroll
    for (int v = 0; v < 8; ++v) {
      const int m = v + (hi ? 8 : 0);
      distS[wave][m * 20 + col]        = fmaf(-2.0f, s0[v], qr0[v] + pn);
      distS[wave][(16 + m) * 20 + col] = fmaf(-2.0f, s1[v], qr1[v] + pn);
    }
    __builtin_amdgcn_wave_barrier();   // store -> cross-lane load, same wave

    // ordered ballquery scan: lane owns query `lane`, db points j0..j0+15.
    // Row base = lane*80 bytes -> 16B aligned: 4x ds_load_b128.
    const float4 r0v = row[0];
    const float4 r1v = row[1];
    const float4 r2v = row[2];
    const float4 r3v = row[3];
    const float d16[16] = {r0v.x, r0v.y, r0v.z, r0v.w, r1v.x, r1v.y, r1v.z, r1v.w,
                           r2v.x, r2v.y, r2v.z, r2v.w, r3v.x, r3v.y, r3v.z, r3v.w};
    float mn = fminf(fminf(fminf(d16[0], d16[1]), fminf(d16[2], d16[3])),
                     fminf(fminf(d16[4], d16[5]), fminf(d16[6], d16[7])));
    mn = fminf(mn, fminf(fminf(fminf(d16[8], d16[9]), fminf(d16[10], d16[11])),
                         fminf(fminf(d16[12], d16[13]), fminf(d16[14], d16[15]))));
    if (mn <= R2_REP && cnt < NS_REP) {      // rare path (~1% of tiles)
#pragma unroll
      for (int n = 0; n < 16; ++n) {
        const float d = d16[n];
        if (cnt < NS_REP && d <= R2_REP) {
          if (cnt == 0) firstd = d;
          insert5(d, best);
          ++cnt;
        }
      }
    }
    __builtin_amdgcn_wave_barrier();   // scan loads before next-iter stores
  }

  // ballquery pads missing slots with the first hit (self always hits -> cnt>=1)
  int npad = NS_REP - cnt;
  if (npad > 5) npad = 5;
  for (int k = 0; k < npad; ++k) insert5(firstd, best);

  // drop smallest (self), relu, sqrt, kernel weight
  const float invh2 = 1.0f / (0.03f * 0.03f);
  float lsum = 0.0f;
#pragma unroll
  for (int k = 1; k < 5; ++k) {
    const float d5 = fmaxf(best[k], 0.0f);
    const float ds = sqrtf(d5);
    const float w  = __expf(-d5 * invh2);
    lsum += 0.07f - ds * w;
  }
#pragma unroll
  for (int off = 16; off; off >>= 1) lsum += __shfl_down(lsum, off);
  if (lane == 0)
    atomicAdd(&out[1], lsum * (1.0f / (float)(N_BATCH * N_PTS * 4)));
}

// ---------------------------------------------------------------------------
extern "C" void kernel_launch(void* const* d_in, const int* in_sizes, int n_in,
                              void* d_out, int out_size, void* d_ws, size_t ws_size,
                              hipStream_t stream) {
  (void)in_sizes; (void)n_in; (void)out_size; (void)ws_size;
  const float* pcd = (const float*)d_in[0];
  float* out = (float*)d_out;
  float* new_xyz = (float*)d_ws;   // 4*409*3 floats

  hipMemsetAsync(out, 0, 2 * sizeof(float), stream);

  fps_kernel<<<N_BATCH, 256, 0, stream>>>(pcd, new_xyz);

  const double kPi = 3.14159265358979323846;
  const double pp[5] = {0.004, 0.008, 0.01, 0.012, 0.016};
  const int G = N_BATCH * NPOINT;
  for (int i = 0; i < 5; ++i) {
    const double p = pp[i];
    const int ns = (int)(8192.0 * p);                 // 32,65,81,98,131
    const double disk = kPi * p / (double)ns;         // pi * r^2 * p / ns (r=1)
    const double expct = sqrt(2.0 * disk / 1.732);
    const float r2 = (float)p;                        // radius^2 = p * 1.0
    const float scale =
        (float)((p * 100.0) * (p * 100.0) /
                ((expct + 1e-8) * (double)G * (double)ns * 5.0));
    const int blocks = (G + 3) / 4;
    uniform_kernel<<<blocks, 128, 0, stream>>>(pcd, new_xyz, out, ns, r2,
                                               (float)expct, scale);
  }

  repulsion_kernel<<<(N_BATCH * N_PTS) / 256, 256, 0, stream>>>(pcd, out);
}